// MicroTransformer_3788161155335
// MI455X (gfx1250) — compile-verified
//
#include <hip/hip_runtime.h>
#include <math.h>

#define DM 5
#define FFD 20
#define NB 16
#define SEQ 512
#define VOCAB 8000
#define NTOK (NB * SEQ)   /* 8192 */

typedef __attribute__((ext_vector_type(2))) float v2f;
typedef __attribute__((ext_vector_type(8))) float v8f;

// ---------------- Stage 1: embedding + (buggy, batch-indexed) positional encoding ----------
__global__ void k_embed_pe(const int* __restrict__ x, const float* __restrict__ emb,
                           float* __restrict__ h) {
    int t = blockIdx.x * blockDim.x + threadIdx.x;
    if (t >= NTOK) return;
    int b = t >> 9;              // batch index (S = 512)
    int tok = x[t];
    const float sqrt5 = 2.2360679774997896f;
    // div = 10000^(-2j/5), j = 0,1,2
    const float d1 = 0.025118864315095794f;   // 10^-1.6
    const float d2 = 0.000630957344480193f;   // 10^-3.2
    float p = (float)b;          // faithful reference bug: pe indexed by batch, bcast over S
    float pe[5];
    pe[0] = __sinf(p);
    pe[1] = __cosf(p);
    pe[2] = __sinf(p * d1);
    pe[3] = __cosf(p * d1);
    pe[4] = __sinf(p * d2);
    const float* e = emb + (size_t)tok * DM;
    float* ho = h + (size_t)t * DM;
#pragma unroll
    for (int d = 0; d < DM; ++d) ho[d] = fmaf(e[d], sqrt5, pe[d]);
}

// ---------------- Stage 2: QKV projection (one thread per (token, out-channel)) ------------
__global__ void k_qkv(const float* __restrict__ h, const float* __restrict__ W,
                      const float* __restrict__ bias, float* __restrict__ qkv) {
    int i = blockIdx.x * blockDim.x + threadIdx.x;   // t*15 + e
    if (i >= NTOK * 15) return;
    int t = i / 15, e = i - t * 15;
    const float* hr = h + (size_t)t * DM;
    const float* wr = W + e * DM;
    float s = bias[e];
#pragma unroll
    for (int d = 0; d < DM; ++d) s = fmaf(hr[d], wr[d], s);
    qkv[i] = s;
}

// ---------------- Stage 3: causal attention, one wave32 per query row ----------------------
__global__ void k_attn(const float* __restrict__ qkv, float* __restrict__ ctx) {
    int wid  = blockIdx.x * (blockDim.x >> 5) + (threadIdx.x >> 5);  // global wave = token
    int lane = threadIdx.x & 31;
    int b  = wid >> 9;
    int qp = wid & 511;
    const float scale = 0.4472135954999579f;   // 1/sqrt(5)
    const float* qv = qkv + (size_t)wid * 15;
    float q0 = qv[0], q1 = qv[1], q2 = qv[2], q3 = qv[3], q4 = qv[4];
    float m = -1e30f, lsum = 0.f;
    float acc[DM] = {0.f, 0.f, 0.f, 0.f, 0.f};
    int base = b << 9;
    for (int k = lane; k <= qp; k += 32) {
        const float* kv = qkv + (size_t)(base + k) * 15 + 5;
        float s = (q0 * kv[0] + q1 * kv[1] + q2 * kv[2] + q3 * kv[3] + q4 * kv[4]) * scale;
        float mn = fmaxf(m, s);
        float cf = __expf(m - mn);
        float pp = __expf(s - mn);
        lsum = lsum * cf + pp;
        const float* vv = kv + 5;
#pragma unroll
        for (int d = 0; d < DM; ++d) acc[d] = fmaf(acc[d], cf, pp * vv[d]);
        m = mn;
    }
    // merge online-softmax states across the 32 lanes (wave32!)
#pragma unroll
    for (int off = 16; off > 0; off >>= 1) {
        float om = __shfl_xor(m, off, 32);
        float ol = __shfl_xor(lsum, off, 32);
        float oa[DM];
#pragma unroll
        for (int d = 0; d < DM; ++d) oa[d] = __shfl_xor(acc[d], off, 32);
        float mn = fmaxf(m, om);
        float c1 = __expf(m - mn), c2 = __expf(om - mn);
        lsum = lsum * c1 + ol * c2;
#pragma unroll
        for (int d = 0; d < DM; ++d) acc[d] = acc[d] * c1 + oa[d] * c2;
        m = mn;
    }
    if (lane == 0) {
        float inv = 1.f / lsum;
        float* co = ctx + (size_t)wid * DM;
#pragma unroll
        for (int d = 0; d < DM; ++d) co[d] = acc[d] * inv;
    }
}

// ---------------- Stage 4: out-proj + residual + LayerNorm1 --------------------------------
__global__ void k_oproj_ln(float* __restrict__ h, const float* __restrict__ ctx,
                           const float* __restrict__ W, const float* __restrict__ bias,
                           const float* __restrict__ g, const float* __restrict__ be) {
    int t = blockIdx.x * blockDim.x + threadIdx.x;
    if (t >= NTOK) return;
    const float* cr = ctx + (size_t)t * DM;
    float* hr = h + (size_t)t * DM;
    float r[DM];
#pragma unroll
    for (int d = 0; d < DM; ++d) {
        float s = bias[d];
#pragma unroll
        for (int e = 0; e < DM; ++e) s = fmaf(cr[e], W[d * DM + e], s);
        r[d] = hr[d] + s;
    }
    float mu = 0.f;
#pragma unroll
    for (int d = 0; d < DM; ++d) mu += r[d];
    mu *= 0.2f;
    float var = 0.f;
#pragma unroll
    for (int d = 0; d < DM; ++d) { float z = r[d] - mu; var = fmaf(z, z, var); }
    var *= 0.2f;
    float inv = rsqrtf(var + 1e-5f);
#pragma unroll
    for (int d = 0; d < DM; ++d) hr[d] = fmaf((r[d] - mu) * inv, g[d], be[d]);
}

// ---------------- Stage 5: FFN (relu) + residual + LayerNorm2 ------------------------------
__global__ void k_ff_ln(float* __restrict__ h,
                        const float* __restrict__ w1, const float* __restrict__ b1,
                        const float* __restrict__ w2, const float* __restrict__ b2,
                        const float* __restrict__ g, const float* __restrict__ be) {
    int t = blockIdx.x * blockDim.x + threadIdx.x;
    if (t >= NTOK) return;
    float* hr = h + (size_t)t * DM;
    float hv[DM];
#pragma unroll
    for (int d = 0; d < DM; ++d) hv[d] = hr[d];
    float y[DM];
#pragma unroll
    for (int d = 0; d < DM; ++d) y[d] = b2[d];
    for (int j = 0; j < FFD; ++j) {
        float f = b1[j];
        const float* wr = w1 + j * DM;
#pragma unroll
        for (int d = 0; d < DM; ++d) f = fmaf(hv[d], wr[d], f);
        f = fmaxf(f, 0.f);
#pragma unroll
        for (int d = 0; d < DM; ++d) y[d] = fmaf(f, w2[d * FFD + j], y[d]);
    }
    float r[DM];
#pragma unroll
    for (int d = 0; d < DM; ++d) r[d] = hv[d] + y[d];
    float mu = 0.f;
#pragma unroll
    for (int d = 0; d < DM; ++d) mu += r[d];
    mu *= 0.2f;
    float var = 0.f;
#pragma unroll
    for (int d = 0; d < DM; ++d) { float z = r[d] - mu; var = fmaf(z, z, var); }
    var *= 0.2f;
    float inv = rsqrtf(var + 1e-5f);
#pragma unroll
    for (int d = 0; d < DM; ++d) hr[d] = fmaf((r[d] - mu) * inv, g[d], be[d]);
}

// ---------------- Stage 6: vocab projection via V_WMMA_F32_16X16X4_F32 ---------------------
// One wave32 computes a 16(token) x 16(vocab) tile: D = A(16x4) x B(4x16) + C, K=5 padded
// to 8 -> two WMMA ops. Store-stream bound (262 MB of logits).
__global__ void k_logits_wmma(const float* __restrict__ h, const float* __restrict__ fcw,
                              const float* __restrict__ fcb, float* __restrict__ out) {
    int mtile = blockIdx.x;                                   // 0..511 (token tiles)
    int ntile = blockIdx.y * 4 + (threadIdx.x >> 5);          // 0..499 (vocab tiles)
    int lane  = threadIdx.x & 31;
    int mn    = lane & 15;
    int hiK   = (lane >> 4) << 1;                             // 0 for lanes 0-15, 2 for 16-31

    // A fragment (16x4, 32-bit): lanes 0-15 -> K={0,1}, lanes 16-31 -> K={2,3}
    const float* hrow = h + (size_t)(mtile * 16 + mn) * DM;
    // B fragment (4x16): lanes 0-15 hold N=0..15 for K={0,1}; lanes 16-31 for K={2,3}
    const float* wrow = fcw + (size_t)(ntile * 16 + mn) * DM;

    v2f a0, b0, a1, b1;
    a0.x = hrow[hiK];         a0.y = hrow[hiK + 1];           // K = 0..3 (all valid)
    b0.x = wrow[hiK];         b0.y = wrow[hiK + 1];
    float h4 = hrow[4], w4 = wrow[4];                         // second step: K=4 valid, 5..7 pad
    a1.x = (lane < 16) ? h4 : 0.f;   a1.y = 0.f;
    b1.x = (lane < 16) ? w4 : 0.f;   b1.y = 0.f;

    v8f c = {};
    c = __builtin_amdgcn_wmma_f32_16x16x4_f32(false, a0, false, b0, (short)0, c, false, false);
    c = __builtin_amdgcn_wmma_f32_16x16x4_f32(false, a1, false, b1, (short)0, c, false, false);

    // D layout: VGPR r, lanes 0-15 -> M=r, N=lane; lanes 16-31 -> M=r+8, N=lane-16
    float bias = fcb[ntile * 16 + mn];
    size_t mbase = (size_t)(mtile * 16 + ((lane >> 4) << 3));
    float* orow = out + mbase * VOCAB + (size_t)(ntile * 16 + mn);
#pragma unroll
    for (int r = 0; r < 8; ++r) orow[(size_t)r * VOCAB] = c[r] + bias;
}

// ---------------- Host launcher ------------------------------------------------------------
extern "C" void kernel_launch(void* const* d_in, const int* in_sizes, int n_in,
                              void* d_out, int out_size, void* d_ws, size_t ws_size,
                              hipStream_t stream) {
    const int*   x    = (const int*)  d_in[0];
    const float* emb  = (const float*)d_in[1];
    const float* ipw  = (const float*)d_in[2];   // [2,15,5]
    const float* ipb  = (const float*)d_in[3];   // [2,15]
    const float* opw  = (const float*)d_in[4];   // [2,5,5]
    const float* opb  = (const float*)d_in[5];   // [2,5]
    const float* ln1g = (const float*)d_in[6];
    const float* ln1b = (const float*)d_in[7];
    const float* ln2g = (const float*)d_in[8];
    const float* ln2b = (const float*)d_in[9];
    const float* ff1w = (const float*)d_in[10];  // [2,20,5]
    const float* ff1b = (const float*)d_in[11];  // [2,20]
    const float* ff2w = (const float*)d_in[12];  // [2,5,20]
    const float* ff2b = (const float*)d_in[13];  // [2,5]
    const float* fcw  = (const float*)d_in[14];  // [8000,5]
    const float* fcb  = (const float*)d_in[15];  // [8000]
    float* out = (float*)d_out;

    float* h   = (float*)d_ws;                   // 40960 f
    float* qkv = h + (size_t)NTOK * DM;          // 122880 f
    float* ctx = qkv + (size_t)NTOK * 15;        // 40960 f

    k_embed_pe<<<(NTOK + 255) / 256, 256, 0, stream>>>(x, emb, h);

    for (int l = 0; l < 2; ++l) {
        k_qkv<<<(NTOK * 15) / 256, 256, 0, stream>>>(h, ipw + l * 75, ipb + l * 15, qkv);
        k_attn<<<NTOK / 8, 256, 0, stream>>>(qkv, ctx);
        k_oproj_ln<<<NTOK / 256, 256, 0, stream>>>(h, ctx, opw + l * 25, opb + l * 5,
                                                   ln1g + l * 5, ln1b + l * 5);
        k_ff_ln<<<NTOK / 256, 256, 0, stream>>>(h, ff1w + l * 100, ff1b + l * 20,
                                                ff2w + l * 100, ff2b + l * 5,
                                                ln2g + l * 5, ln2b + l * 5);
    }

    dim3 grid(NTOK / 16, VOCAB / 16 / 4);        // (512, 125)
    k_logits_wmma<<<grid, 128, 0, stream>>>(h, fcw, fcb, out);
}